// EntmaxAttention_34488587387514
// MI455X (gfx1250) — compile-verified
//
#include <hip/hip_runtime.h>

typedef __attribute__((ext_vector_type(2))) float v2f;
typedef __attribute__((ext_vector_type(8))) float v8f;

#define WMMA_F32(a, b, c) \
  __builtin_amdgcn_wmma_f32_16x16x4_f32(false, (a), false, (b), (short)0, (c), false, false)

namespace {
constexpr int S = 2048;          // sequence length
constexpr int D = 64;            // head dim
constexpr int LDSTRIDE = 2052;   // floats; 2048+4 kills LDS bank conflicts, keeps 16B align
constexpr float INV_T = 0.125f;  // 1 / TEMPERATURE
}

__global__ __launch_bounds__(256)
void sparsemax_attn_kernel(const float* __restrict__ Qg,
                           const float* __restrict__ Kg,
                           const float* __restrict__ Vg,
                           float* __restrict__ Og)
{
  // 16 x 2048 fp32 score tile (128.25 KB) + 4 KB partial-sum scratch.
  __shared__ float sc[16 * LDSTRIDE];
  __shared__ float psum[4 * 256];

  const int wg    = blockIdx.x;     // 0 .. B*H*(S/16)-1
  const int qtile = wg & 127;       // S/16 = 128 tiles
  const int bh    = wg >> 7;        // 0 .. 31
  const int tid   = threadIdx.x;
  const int wave  = tid >> 5;       // 0 .. 7 (wave32)
  const int lane  = tid & 31;
  const int lhalf = lane >> 4;      // 0 | 1 (half-wave for WMMA layouts)
  const int lmod  = lane & 15;

  const float* Q = Qg + (size_t)bh * S * D + (size_t)qtile * 16 * D;
  const float* K = Kg + (size_t)bh * S * D;
  const float* V = Vg + (size_t)bh * S * D;
  float*       O = Og + (size_t)bh * S * D + (size_t)qtile * 16 * D;

  // ================= GEMM1: scores = (Q * K^T) / T  -> LDS =================
  // A fragments (16x64 Q tile): lane l%16 = row M, VGPR pair holds K=k0+2*(l/16)+{0,1}.
  v2f afrag[16];
  #pragma unroll
  for (int ks = 0; ks < 16; ++ks) {
    v2f t = *(const v2f*)(Q + lmod * D + ks * 4 + 2 * lhalf);
    t.x *= INV_T;
    t.y *= INV_T;
    afrag[ks] = t;   // pre-scale A once instead of scaling every C tile
  }

  // 128 N-tiles of 16 keys each; 16 tiles per wave.
  #pragma unroll 1
  for (int t = 0; t < 16; ++t) {
    const int nt = wave * 16 + t;
    // B[kk, n] = K[n_global, kk]; contiguous float2 along D.
    const float* kb = K + (size_t)(nt * 16 + lmod) * D + 2 * lhalf;
    v8f c = {};
    #pragma unroll
    for (int ks = 0; ks < 16; ++ks) {
      v2f b = *(const v2f*)(kb + ks * 4);
      c = WMMA_F32(afrag[ks], b, c);
    }
    // C layout: lanes 0-15 -> M=r, lanes 16-31 -> M=r+8; N = lane%16.
    float* sp = sc + lhalf * 8 * LDSTRIDE + nt * 16 + lmod;
    #pragma unroll
    for (int r = 0; r < 8; ++r) sp[r * LDSTRIDE] = c[r];
  }
  __syncthreads();

  // ================= sparsemax per row (Michelot iteration) =================
  // Each wave owns rows 2*wave and 2*wave+1; row cached in 64 VGPRs per lane.
  #pragma unroll 1
  for (int rr = 0; rr < 2; ++rr) {
    float* rp = sc + (wave * 2 + rr) * LDSTRIDE;
    float x[64];
    float s = 0.0f;
    #pragma unroll
    for (int i = 0; i < 64; ++i) {
      x[i] = rp[i * 32 + lane];   // stride-32: conflict-free
      s += x[i];
    }
    #pragma unroll
    for (int off = 16; off > 0; off >>= 1) s += __shfl_xor(s, off, 32);

    // tau_0 = (sum-1)/n <= tau*; iteration is monotone and idempotent at the
    // fixed point, so a fixed iteration cap is safe. Early-exit is wave-uniform.
    float tau = (s - 1.0f) * (1.0f / (float)S);
    #pragma unroll 1
    for (int it = 0; it < 32; ++it) {
      float ls = 0.0f, lk = 0.0f;
      #pragma unroll
      for (int i = 0; i < 64; ++i) {
        const bool act = x[i] > tau;
        ls += act ? x[i] : 0.0f;
        lk += act ? 1.0f : 0.0f;
      }
      #pragma unroll
      for (int off = 16; off > 0; off >>= 1) {
        ls += __shfl_xor(ls, off, 32);
        lk += __shfl_xor(lk, off, 32);
      }
      const float ntau = (ls - 1.0f) / lk;   // lk >= 1 always (max elem stays active)
      if (ntau == tau) break;                // uniform across the wave
      tau = ntau;
    }

    #pragma unroll
    for (int i = 0; i < 64; ++i) {
      const float a = x[i] - tau;
      rp[i * 32 + lane] = a > 0.0f ? a : 0.0f;   // attn weights back to LDS
    }
  }
  __syncthreads();

  // ================= GEMM2: out = attn * V =================
  // 4 D-tiles x 2 K-halves over 8 waves; partials combined through LDS.
  const int dtile = wave & 3;
  const int khalf = wave >> 2;
  const int kbase = khalf * (S / 2);

  v8f c2 = {};
  const float* vb = V + (size_t)(kbase + 2 * lhalf) * D + dtile * 16 + lmod;
  const float* ab = sc + lmod * LDSTRIDE + kbase + 2 * lhalf;
  #pragma unroll 4
  for (int k0 = 0; k0 < S / 2; k0 += 4) {
    v2f a = *(const v2f*)(ab + k0);          // attn A-fragment from LDS (8B aligned)
    const float* pv = vb + (size_t)k0 * D;   // V B-fragment: two rows, stride D
    v2f b;
    b.x = pv[0];
    b.y = pv[D];
    c2 = WMMA_F32(a, b, c2);
  }

  if (khalf == 1) {
    float* pp = psum + dtile * 256 + lane;
    #pragma unroll
    for (int r = 0; r < 8; ++r) pp[r * 32] = c2[r];
  }
  __syncthreads();
  if (khalf == 0) {
    const float* pp = psum + dtile * 256 + lane;
    #pragma unroll
    for (int r = 0; r < 8; ++r) {
      const float val = c2[r] + pp[r * 32];
      O[(size_t)(r + lhalf * 8) * D + dtile * 16 + lmod] = val;
    }
  }
}

extern "C" void kernel_launch(void* const* d_in, const int* in_sizes, int n_in,
                              void* d_out, int out_size, void* d_ws, size_t ws_size,
                              hipStream_t stream) {
  (void)in_sizes; (void)n_in; (void)out_size; (void)d_ws; (void)ws_size;
  const float* q = (const float*)d_in[0];
  const float* k = (const float*)d_in[1];
  const float* v = (const float*)d_in[2];
  float* out = (float*)d_out;

  // grid = B*H*(S/16) = 2*16*128 = 4096 workgroups, 256 threads (8 wave32s).
  sparsemax_attn_kernel<<<dim3(4096), dim3(256), 0, stream>>>(q, k, v, out);
}